// MPSDist_56891136803227
// MI455X (gfx1250) — compile-verified
//
#include <hip/hip_runtime.h>
#include <stdint.h>

#define RANK   64
#define SEQLEN 256
#define VOCABN 32768

typedef int   v4i __attribute__((ext_vector_type(4)));
typedef int   v8i __attribute__((ext_vector_type(8)));
typedef float v2f __attribute__((ext_vector_type(2)));
typedef float v8f __attribute__((ext_vector_type(8)));

// ---------------------------------------------------------------------------
// TDM: DMA one 64x64 f32 tile (row-major, row stride = VOCAB*RANK elements)
// from global memory into LDS.  D# layout per CDNA5 ISA.
//   group0: [1:0]=count=1, [63:32]=lds byte addr, [120:64]=global byte addr,
//           [127:126]=type=2
//   group1: data_size=2 (4B), tensor_dim0=VOCAB*RANK, tensor_dim1=64,
//           tile_dim0=64, tile_dim1=64, tensor_dim0_stride=VOCAB*RANK
// ---------------------------------------------------------------------------
__device__ __forceinline__ void tdm_load_tile_64x64_f32(unsigned lds_byte,
                                                        unsigned long long gaddr)
{
    v4i g0;
    g0.x = 1;                                   // count=1, user descriptor
    g0.y = (int)lds_byte;                       // LDS byte address
    g0.z = (int)(uint32_t)gaddr;                // global addr [31:0]
    g0.w = (int)(((uint32_t)(gaddr >> 32)) | 0x80000000u); // [56:32] | type=2

    v8i g1;
    g1[0] = 0x00020000;                         // data_size=2 (4 bytes)
    g1[1] = 0;                                  // tensor_dim0[15:0]=0 (of 0x200000)
    g1[2] = (int)((64u << 16) | 0x20u);         // tensor_dim0[31:16]=0x20 | tensor_dim1 lo=64
    g1[3] = (int)(64u << 16);                   // tensor_dim1 hi=0 | tile_dim0=64
    g1[4] = 64;                                 // tile_dim1=64, tile_dim2=0
    g1[5] = 0x00200000;                         // tensor_dim0_stride = VOCAB*RANK = 2097152
    g1[6] = 0;                                  // stride hi / dim1_stride lo
    g1[7] = 0;

#if __has_builtin(__builtin_amdgcn_tensor_load_to_lds_d2)
    __builtin_amdgcn_tensor_load_to_lds_d2(g0, g1, 0);
#elif __has_builtin(__builtin_amdgcn_tensor_load_to_lds)
    v4i z4 = {0, 0, 0, 0};
 #if defined(__clang_major__) && __clang_major__ >= 23
    v8i z8 = {0, 0, 0, 0, 0, 0, 0, 0};
    __builtin_amdgcn_tensor_load_to_lds(g0, g1, z4, z4, z8, 0);
 #else
    __builtin_amdgcn_tensor_load_to_lds(g0, g1, z4, z4, 0);
 #endif
#else
    // Host compilation pass (no AMDGCN builtins): device never takes this path.
    (void)g0; (void)g1;
#endif
}

// ---------------------------------------------------------------------------
// Kernel 1: one wave32 per batch row.  256 sequential steps, TDM double
// buffering of the 16KB per-token matrix, VALU f32 recurrence from LDS.
// Lane j owns output components s = 2j, 2j+1.
// ---------------------------------------------------------------------------
__global__ __launch_bounds__(32)
void mps_chain_kernel(const int*   __restrict__ y,
                      const float* __restrict__ alpha,
                      const float* __restrict__ beta,
                      const float* __restrict__ core,
                      float*       __restrict__ lT,     // [batch][RANK] or null
                      float*       __restrict__ out,
                      int fuse_dot)
{
    __shared__ __align__(16) float tile[2][RANK * RANK]; // 32 KB double buffer
    __shared__ __align__(16) float lsh[RANK];            // left boundary vector
    __shared__ int ytok[SEQLEN];

    const int b = blockIdx.x;
    const int j = threadIdx.x; // 0..31

    // stage this row's tokens into LDS (8 per lane)
    for (int k = j; k < SEQLEN; k += 32)
        ytok[k] = y[(size_t)b * SEQLEN + k];

    // l0 = alpha^2
    float ax = alpha[2 * j], ay = alpha[2 * j + 1];
    float2 l;
    l.x = ax * ax;
    l.y = ay * ay;
    *(float2*)&lsh[2 * j] = l;
    __syncthreads();

    const unsigned long long cbase = (unsigned long long)(uintptr_t)core;
    const unsigned t0 = (unsigned)(uintptr_t)&tile[0][0]; // LDS byte offsets
    const unsigned t1 = (unsigned)(uintptr_t)&tile[1][0];

    // prefetch matrix for t=0
    {
        int tok = __builtin_amdgcn_readfirstlane(ytok[0]);
        tdm_load_tile_64x64_f32(
            t0, cbase + (unsigned long long)(uint32_t)tok * (RANK * sizeof(float)));
    }

    int buf = 0;
    for (int t = 0; t < SEQLEN; ++t) {
        if (t + 1 < SEQLEN) {
            int tok = __builtin_amdgcn_readfirstlane(ytok[t + 1]);
            // all DS reads of the buffer about to be overwritten must drain
            // before the TDM engine can write it
            asm volatile("s_wait_dscnt 0x0" ::: "memory");
            tdm_load_tile_64x64_f32(
                buf ? t0 : t1,
                cbase + (unsigned long long)(uint32_t)tok * (RANK * sizeof(float)));
            // TDM completes in-order per wave: cnt<=1 => tile[buf] is resident
            asm volatile("s_wait_tensorcnt 0x1" ::: "memory");
        } else {
            asm volatile("s_wait_tensorcnt 0x0" ::: "memory");
        }

        const float* tl = &tile[buf][0];
        float2 acc;
        acc.x = 0.0f;
        acc.y = 0.0f;
        #pragma unroll
        for (int r4 = 0; r4 < RANK; r4 += 4) {
            // uniform address -> LDS broadcast of 4 l-components
            float4 lv = *(const float4*)&lsh[r4];
            float2 m0 = *(const float2*)&tl[(r4 + 0) * RANK + 2 * j];
            float2 m1 = *(const float2*)&tl[(r4 + 1) * RANK + 2 * j];
            float2 m2 = *(const float2*)&tl[(r4 + 2) * RANK + 2 * j];
            float2 m3 = *(const float2*)&tl[(r4 + 3) * RANK + 2 * j];
            // core is squared on the fly (reference uses core**2)
            acc.x = fmaf(lv.x, m0.x * m0.x, acc.x);
            acc.y = fmaf(lv.x, m0.y * m0.y, acc.y);
            acc.x = fmaf(lv.y, m1.x * m1.x, acc.x);
            acc.y = fmaf(lv.y, m1.y * m1.y, acc.y);
            acc.x = fmaf(lv.z, m2.x * m2.x, acc.x);
            acc.y = fmaf(lv.z, m2.y * m2.y, acc.y);
            acc.x = fmaf(lv.w, m3.x * m3.x, acc.x);
            acc.y = fmaf(lv.w, m3.y * m3.y, acc.y);
        }
        // publish l for the next step (same-wave LDS ops stay in order)
        *(float2*)&lsh[2 * j] = acc;
        l = acc;
        buf ^= 1;
    }

    if (fuse_dot) {
        float bx = beta[2 * j], by = beta[2 * j + 1];
        float partial = l.x * bx * bx + l.y * by * by;
        for (int off = 16; off > 0; off >>= 1)
            partial += __shfl_xor(partial, off, 32);
        if (j == 0) out[b] = partial;
    } else {
        *(float2*)&lT[(size_t)b * RANK + 2 * j] = l;
    }
}

// ---------------------------------------------------------------------------
// Kernel 2: out[b] = sum_s L[b,s] * beta[s]^2 for 16 batches per wave via
// V_WMMA_F32_16X16X4_F32.  A(16x4) = 16 l-vectors (r-chunk), B(4x16) = beta^2
// replicated across N, accumulate over 16 r-chunks.  D[i,j] == out[i] for all j.
// ---------------------------------------------------------------------------
__global__ __launch_bounds__(32)
void mps_dot_wmma_kernel(const float* __restrict__ lT,
                         const float* __restrict__ beta,
                         float*       __restrict__ out,
                         int nbatch)
{
#if __has_builtin(__builtin_amdgcn_wmma_f32_16x16x4_f32)
    const int g    = blockIdx.x;    // group of 16 batch rows
    const int lane = threadIdx.x;   // 0..31
    const int m    = lane & 15;     // batch row within group (A M-index)
    const int kh   = lane >> 4;     // K half: lanes 0-15 -> K=0,1; 16-31 -> K=2,3

    v8f acc = {0, 0, 0, 0, 0, 0, 0, 0};
    #pragma unroll
    for (int c = 0; c < RANK / 4; ++c) {
        int k0 = 4 * c + 2 * kh;
        float2 a2 = *(const float2*)&lT[((size_t)g * 16 + m) * RANK + k0];
        float b0 = beta[k0];     b0 *= b0;
        float b1 = beta[k0 + 1]; b1 *= b1;
        v2f A; A.x = a2.x; A.y = a2.y;
        v2f B; B.x = b0;   B.y = b1;
        acc = __builtin_amdgcn_wmma_f32_16x16x4_f32(
            false, A, false, B, (short)0, acc, false, false);
    }
    // D layout: lanes 0-15 hold M=v (VGPR v), lanes 16-31 hold M=v+8.
    // Column N=0 of each half-wave carries all 16 results.
    if (m == 0) {
        int bb = g * 16 + kh * 8;
        #pragma unroll
        for (int v = 0; v < 8; ++v)
            if (bb + v < nbatch) out[bb + v] = acc[v];
    }
#else
    int idx = blockIdx.x * 32 + threadIdx.x;
    if (idx < nbatch) {
        float s = 0.0f;
        for (int r = 0; r < RANK; ++r) {
            float bv = beta[r];
            s += lT[(size_t)idx * RANK + r] * bv * bv;
        }
        out[idx] = s;
    }
#endif
}

// ---------------------------------------------------------------------------
extern "C" void kernel_launch(void* const* d_in, const int* in_sizes, int n_in,
                              void* d_out, int out_size, void* d_ws, size_t ws_size,
                              hipStream_t stream)
{
    (void)n_in; (void)out_size;
    const int*   y     = (const int*)  d_in[0];
    const float* alpha = (const float*)d_in[1];
    const float* beta  = (const float*)d_in[2];
    const float* core  = (const float*)d_in[3];
    float* out = (float*)d_out;

    const int batch = in_sizes[0] / SEQLEN;
    const size_t need = (size_t)batch * RANK * sizeof(float);

    if (ws_size >= need && (batch % 16) == 0) {
        float* lT = (float*)d_ws;
        mps_chain_kernel<<<batch, 32, 0, stream>>>(y, alpha, beta, core, lT, out, 0);
        mps_dot_wmma_kernel<<<batch / 16, 32, 0, stream>>>(lT, beta, out, batch);
    } else {
        mps_chain_kernel<<<batch, 32, 0, stream>>>(y, alpha, beta, core, nullptr, out, 1);
    }
}